// DownwardPropagation_55276229099594
// MI455X (gfx1250) — compile-verified
//
#include <hip/hip_runtime.h>
#include <stdint.h>

#define B_TOT 16384
#define L_LAY 60
#define C_CH  48
#define WPB   8                        // waves per block
#define BLOCK (WPB * 32)
#define BUF_FLOATS  (8 * C_CH)         // 384 floats: one layer's 8 property rows
#define WAVE_FLOATS (2 * BUF_FLOATS)   // double buffered

typedef int v4i __attribute__((vector_size(16)));
typedef __attribute__((address_space(1))) v4i gv4i;   // global int4
typedef __attribute__((address_space(3))) v4i lv4i;   // LDS int4

__device__ __forceinline__ float wave_sum(float v) {
#pragma unroll
  for (int m = 16; m > 0; m >>= 1) v += __shfl_xor(v, m, 32);
  return v;
}

__global__ __launch_bounds__(BLOCK) void radprop_kernel(
    const float* __restrict__ p_td,  const float* __restrict__ p_tdf,
    const float* __restrict__ p_tmd, const float* __restrict__ p_tmf,
    const float* __restrict__ p_rbd, const float* __restrict__ p_rbf,
    const float* __restrict__ p_atd, const float* __restrict__ p_atf,
    const float* __restrict__ p_rmd, const float* __restrict__ p_fd0,
    const float* __restrict__ p_ff0, float* __restrict__ out) {
  __shared__ __align__(16) float smem[WPB * WAVE_FLOATS];

  const int lane = threadIdx.x & 31;
  const int wid  = threadIdx.x >> 5;
  const int b    = blockIdx.x * WPB + wid;

  // ---- async staging setup: 8 arrays * 12 chunks(16B) = 96 slots = 3 full-wave b128 ops
  uint64_t gsrc[3];
  unsigned ldst[3];
  {
    const unsigned smem_base = (unsigned)(uintptr_t)(&smem[0]);   // low 32 bits = LDS offset
    const unsigned wave_base = smem_base + (unsigned)(wid * WAVE_FLOATS * 4);
#pragma unroll
    for (int i = 0; i < 3; ++i) {
      const int slot = i * 32 + lane;
      const int a    = slot / 12;   // which property array
      const int ch   = slot % 12;   // 16B chunk within the 192B row
      uint64_t p = (uint64_t)(uintptr_t)p_td;
      p = (a == 1) ? (uint64_t)(uintptr_t)p_tdf : p;
      p = (a == 2) ? (uint64_t)(uintptr_t)p_tmd : p;
      p = (a == 3) ? (uint64_t)(uintptr_t)p_tmf : p;
      p = (a == 4) ? (uint64_t)(uintptr_t)p_rbd : p;
      p = (a == 5) ? (uint64_t)(uintptr_t)p_rbf : p;
      p = (a == 6) ? (uint64_t)(uintptr_t)p_atd : p;
      p = (a == 7) ? (uint64_t)(uintptr_t)p_atf : p;
      gsrc[i] = p + (uint64_t)b * (L_LAY * C_CH * 4) + (unsigned)(ch * 16);
      ldst[i] = wave_base + (unsigned)(a * C_CH * 4 + ch * 16);
    }
  }

  // each call consumes one layer and advances; caller picks the LDS buffer
  auto issue = [&](int buf) {
    const unsigned boff = (unsigned)(buf * BUF_FLOATS * 4);
#pragma unroll
    for (int i = 0; i < 3; ++i) {
      // async copy, non-temporal (cpol TH=NT=1): 1.5GB streamed once, keep it out of L2
      __builtin_amdgcn_global_load_async_to_lds_b128(
          (gv4i*)(uintptr_t)gsrc[i],
          (lv4i*)(uintptr_t)(ldst[i] + boff),
          0, 1);
      gsrc[i] += C_CH * 4;   // advance one layer (192B)
    }
  };

  // ---- prefetch pipeline depth 2: layers 0 and 1 in flight before compute starts
  issue(0);
  issue(1);

  // ---- initial (top-of-atmosphere) fluxes; lane owns channel c0, and c1 if lane<16
  const bool hi = lane < 16;
  const int  c0 = lane;
  const int  c1 = hi ? lane + 32 : 0;
  const size_t fb = (size_t)b * C_CH;
  float fd_0 = __builtin_nontemporal_load(&p_fd0[fb + c0]);
  float ff_0 = __builtin_nontemporal_load(&p_ff0[fb + c0]);
  float rm_0 = __builtin_nontemporal_load(&p_rmd[fb + c0]);
  float fd_1 = 0.f, ff_1 = 0.f, rm_1 = 0.f;
  if (hi) {
    fd_1 = __builtin_nontemporal_load(&p_fd0[fb + c1]);
    ff_1 = __builtin_nontemporal_load(&p_ff0[fb + c1]);
    rm_1 = __builtin_nontemporal_load(&p_rmd[fb + c1]);
  }

  float* out_fdd = out;
  float* out_fdf = out + (size_t)B_TOT * (L_LAY + 1);
  float* out_fup = out + (size_t)2 * B_TOT * (L_LAY + 1);
  float* out_abs = out + (size_t)3 * B_TOT * (L_LAY + 1);
  const size_t ob = (size_t)b * (L_LAY + 1);

  {
    float s_fd = wave_sum(fd_0 + fd_1);
    float s_ff = wave_sum(ff_0 + ff_1);
    float s_up = wave_sum(fd_0 * rm_0 + fd_1 * rm_1);
    if (lane == 0) {
      __builtin_nontemporal_store(s_fd, &out_fdd[ob]);
      __builtin_nontemporal_store(s_ff, &out_fdf[ob]);
      __builtin_nontemporal_store(s_up, &out_fup[ob]);
    }
  }

  const float* wbuf = &smem[wid * WAVE_FLOATS];

  // computes layer update from `buf`, leaves unreduced per-lane partials in p*
  auto layer_partials = [&](const float* buf, float& pfd, float& pff,
                            float& pup, float& pab) {
    const float td  = buf[0 * C_CH + c0];
    const float tdf = buf[1 * C_CH + c0];
    const float tmd = buf[2 * C_CH + c0];
    const float tmf = buf[3 * C_CH + c0];
    const float rbd = buf[4 * C_CH + c0];
    const float rbf = buf[5 * C_CH + c0];
    const float atd = buf[6 * C_CH + c0];
    const float atf = buf[7 * C_CH + c0];

    float ab0 = fd_0 * atd + ff_0 * atf;
    float up0 = fd_0 * rbd + ff_0 * rbf;
    float nff = fd_0 * tmd + ff_0 * (tdf + tmf);
    fd_0 = fd_0 * td;
    ff_0 = nff;

    float ab1 = 0.f, up1 = 0.f;
    if (hi) {
      const float td_  = buf[0 * C_CH + c1];
      const float tdf_ = buf[1 * C_CH + c1];
      const float tmd_ = buf[2 * C_CH + c1];
      const float tmf_ = buf[3 * C_CH + c1];
      const float rbd_ = buf[4 * C_CH + c1];
      const float rbf_ = buf[5 * C_CH + c1];
      const float atd_ = buf[6 * C_CH + c1];
      const float atf_ = buf[7 * C_CH + c1];
      ab1 = fd_1 * atd_ + ff_1 * atf_;
      up1 = fd_1 * rbd_ + ff_1 * rbf_;
      float nff1 = fd_1 * tmd_ + ff_1 * (tdf_ + tmf_);
      fd_1 = fd_1 * td_;
      ff_1 = nff1;
    }
    pfd = fd_0 + fd_1;
    pff = ff_0 + ff_1;
    pup = up0 + up1;
    pab = ab0 + ab1;
  };

  auto reduce_store = [&](int l, float pfd, float pff, float pup, float pab) {
    float s_fd = wave_sum(pfd);
    float s_ff = wave_sum(pff);
    float s_up = wave_sum(pup);
    float s_ab = wave_sum(pab);
    if (lane == 0) {
      __builtin_nontemporal_store(s_fd, &out_fdd[ob + l + 1]);
      __builtin_nontemporal_store(s_ff, &out_fdf[ob + l + 1]);
      __builtin_nontemporal_store(s_up, &out_fup[ob + l + 1]);
      __builtin_nontemporal_store(s_ab, &out_abs[(size_t)b * L_LAY + l]);
    }
  };

  // ---- branch-free steady state: always one issue per iteration (layers 2..L-1)
#pragma unroll 1
  for (int l = 0; l < L_LAY - 2; ++l) {
    // in-order async completion: <=3 outstanding means layer l has landed
    asm volatile("s_wait_asynccnt 3" ::: "memory");
    float pfd, pff, pup, pab;
    layer_partials(wbuf + (l & 1) * BUF_FLOATS, pfd, pff, pup, pab);
    // our ds_loads of buf[l&1] are already retired (values consumed) -> free fence
    asm volatile("s_wait_dscnt 0" ::: "memory");
    issue(l & 1);   // layer l+2 -> buffer (l+2)&1 == l&1
    reduce_store(l, pfd, pff, pup, pab);
  }
  // peeled layer L-2: one layer still in flight behind it
  {
    asm volatile("s_wait_asynccnt 3" ::: "memory");
    float pfd, pff, pup, pab;
    layer_partials(wbuf + ((L_LAY - 2) & 1) * BUF_FLOATS, pfd, pff, pup, pab);
    reduce_store(L_LAY - 2, pfd, pff, pup, pab);
  }
  // peeled layer L-1: drain
  {
    asm volatile("s_wait_asynccnt 0" ::: "memory");
    float pfd, pff, pup, pab;
    layer_partials(wbuf + ((L_LAY - 1) & 1) * BUF_FLOATS, pfd, pff, pup, pab);
    reduce_store(L_LAY - 1, pfd, pff, pup, pab);
  }
}

extern "C" void kernel_launch(void* const* d_in, const int* in_sizes, int n_in,
                              void* d_out, int out_size, void* d_ws, size_t ws_size,
                              hipStream_t stream) {
  (void)in_sizes; (void)n_in; (void)out_size; (void)d_ws; (void)ws_size;
  const float* td  = (const float*)d_in[0];
  const float* tdf = (const float*)d_in[1];
  const float* tmd = (const float*)d_in[2];
  const float* tmf = (const float*)d_in[3];
  const float* rbd = (const float*)d_in[4];
  const float* rbf = (const float*)d_in[5];
  const float* atd = (const float*)d_in[6];
  const float* atf = (const float*)d_in[7];
  const float* rmd = (const float*)d_in[8];
  const float* fd0 = (const float*)d_in[9];
  const float* ff0 = (const float*)d_in[10];
  float* out = (float*)d_out;

  dim3 grid(B_TOT / WPB);
  dim3 block(BLOCK);
  radprop_kernel<<<grid, block, 0, stream>>>(td, tdf, tmd, tmf, rbd, rbf,
                                             atd, atf, rmd, fd0, ff0, out);
}